// DFMBPSROIAlign_52639119180040
// MI455X (gfx1250) — compile-verified
//
#include <hip/hip_runtime.h>

// Problem constants (match reference)
#define CC     10
#define PHH    7
#define PWW    7
#define HH     34
#define WW     34
#define PP     (PHH * PWW)   // 49
#define HWSZ   (HH * WW)     // 1156 pixels per (c,ph,pw) slice
#define CPAD   12            // channels padded 10 -> 12 floats (48 B, 16B-aligned)
#define RPB    256           // ROIs per block == threads per block (8 wave32s)

// ---- CDNA5 async global->LDS copies (ASYNCcnt path) -----------------------
// Per-lane scatter copy: 4 bytes from the global address in vaddr to the LDS address in vdst. Generic LDS
// pointers carry the DS byte offset in their low 32 bits.
__device__ __forceinline__ void async_cp4(void* lds_dst, const void* gsrc) {
  asm volatile("global_load_async_to_lds_b32 %0, %1, off"
               :: "v"((unsigned)(unsigned long long)(__UINTPTR_TYPE__)lds_dst),
                  "v"((unsigned long long)(__UINTPTR_TYPE__)gsrc)
               : "memory");
}

__device__ __forceinline__ void wait_async0() {
  asm volatile("s_wait_asynccnt 0" ::: "memory");
}
// ---------------------------------------------------------------------------

// Load one pixel's 10 channel values from the interleaved LDS tile:
// 2x ds_load_b128 + 1x ds_load_b64 (record is 48B and 16B-aligned).
__device__ __forceinline__ void load_pix(const float* __restrict__ t, int idx,
                                         float v[CC]) {
  const float4* p4 = reinterpret_cast<const float4*>(t + idx * CPAD);
  const float4 a = p4[0];
  const float4 b = p4[1];
  const float2 c = *reinterpret_cast<const float2*>(t + idx * CPAD + 8);
  v[0] = a.x; v[1] = a.y; v[2] = a.z; v[3] = a.w;
  v[4] = b.x; v[5] = b.y; v[6] = b.z; v[7] = b.w;
  v[8] = c.x; v[9] = c.y;
}

__global__ __launch_bounds__(RPB) void dfmb_psroi_align_kernel(
    const float* __restrict__ ft,    // (C*PH*PW, H, W) contiguous
    const float* __restrict__ rois,  // (N, 5)
    float* __restrict__ out,         // (N, C, P)
    int N) {
  // Channel-interleaved, padded tile for this bin: [y*W+x][c], 55488 B.
  __shared__ __align__(16) float tile[HWSZ * CPAD];

  const int p   = blockIdx.y;        // bin index = ph*PW + pw
  const int tid = threadIdx.x;

  // ---- Stage + transpose via per-lane async B32 scatter copies ------------
  // Global element e = c*HW + yx (slice-major) -> LDS offset (yx*CPAD + c).
  constexpr int TOTE = CC * HWSZ;    // 11560 elements
  for (int e = tid; e < TOTE; e += RPB) {
    const int c  = e / HWSZ;
    const int yx = e - c * HWSZ;
    async_cp4(&tile[yx * CPAD + c],
              ft + (size_t)(c * PP + p) * HWSZ + yx);
  }

  // ---- ROI geometry overlapped with the in-flight DMA ---------------------
  const int n = blockIdx.x * RPB + tid;
  float x1r = 0.f, y1r = 0.f, x2r = 0.f, y2r = 0.f;
  if (n < N) {
    const float inv_stride = 1.0f / 16.0f;  // heat_map_a
    x1r = rois[(size_t)n * 5 + 1] * inv_stride;
    y1r = rois[(size_t)n * 5 + 2] * inv_stride;
    x2r = rois[(size_t)n * 5 + 3] * inv_stride;
    y2r = rois[(size_t)n * 5 + 4] * inv_stride;
  }
  float roi_w = x2r - x1r; roi_w = (roi_w > 0.1f) ? roi_w : 0.1f;  // threshold
  float roi_h = y2r - y1r; roi_h = (roi_h > 0.1f) ? roi_h : 0.1f;

  const int ph = p / PWW;
  const int pw = p - ph * PWW;
  const float bin_h = roi_h * (1.0f / (float)PHH);
  const float bin_w = roi_w * (1.0f / (float)PWW);
  const float sub_h = bin_h * 0.25f;   // bin / sample_per_part(4)
  const float sub_w = bin_w * 0.25f;
  const float hstart = floorf(y1r + (float)ph * bin_h);
  const float wstart = floorf(x1r + (float)pw * bin_w);

  // ---- Complete staging (each wave drains its ASYNCcnt, then WG sync) -----
  wait_async0();
  __syncthreads();

  if (n >= N) return;

  float sum[CC];
#pragma unroll
  for (int c = 0; c < CC; ++c) sum[c] = 0.0f;
  float count = 0.0f;

  for (int ih = 0; ih < 4; ++ih) {
    const float h   = hstart + ((float)ih + 0.5f) * sub_h;
    const float y1f = floorf(h);
    const float y2f = ceilf(h);
    const float y1v = (y1f >= 0.0f && y1f < (float)HH) ? 1.0f : 0.0f;
    const float y2v = (y2f >= 0.0f && y2f < (float)HH) ? 1.0f : 0.0f;
    const int   y1i = (int)fminf(fmaxf(y1f, 0.0f), (float)(HH - 1));
    const int   y2i = (int)fminf(fmaxf(y2f, 0.0f), (float)(HH - 1));
    const float dy  = h - (float)y1i;   // measured vs CLAMPED corner (ref)
    const bool  hok = (h > -1.0f) && (h < (float)HH);

    for (int iw = 0; iw < 4; ++iw) {
      const float w   = wstart + ((float)iw + 0.5f) * sub_w;
      const float x1f = floorf(w);
      const float x2f = ceilf(w);
      const float x1v = (x1f >= 0.0f && x1f < (float)WW) ? 1.0f : 0.0f;
      const float x2v = (x2f >= 0.0f && x2f < (float)WW) ? 1.0f : 0.0f;
      const int   x1i = (int)fminf(fmaxf(x1f, 0.0f), (float)(WW - 1));
      const int   x2i = (int)fminf(fmaxf(x2f, 0.0f), (float)(WW - 1));
      const float dx  = w - (float)x1i;
      const float kf  = (hok && (w > -1.0f) && (w < (float)WW)) ? 1.0f : 0.0f;

      // channel-invariant bilinear weights; validity + keep folded in
      const float w11 = (1.0f - dx) * (1.0f - dy) * (x1v * y1v) * kf;
      const float w12 = (1.0f - dx) * dy          * (x1v * y2v) * kf;
      const float w21 = dx          * (1.0f - dy) * (x2v * y1v) * kf;
      const float w22 = dx          * dy          * (x2v * y2v) * kf;

      const int i11 = y1i * WW + x1i;
      const int i12 = y2i * WW + x1i;
      const int i21 = y1i * WW + x2i;
      const int i22 = y2i * WW + x2i;

      float v11[CC], v12[CC], v21[CC], v22[CC];
      load_pix(tile, i11, v11);
      load_pix(tile, i12, v12);
      load_pix(tile, i21, v21);
      load_pix(tile, i22, v22);

      count += kf;
#pragma unroll
      for (int c = 0; c < CC; ++c) {
        float v = w11 * v11[c];
        v = fmaf(w12, v12[c], v);
        v = fmaf(w21, v21[c], v);
        v = fmaf(w22, v22[c], v);
        sum[c] += v;
      }
    }
  }

  const float denom = (count > 0.0f) ? count : 1.0f;
  const float inv   = 1.0f / denom;
  float* o = out + (size_t)n * (CC * PP) + p;
#pragma unroll
  for (int c = 0; c < CC; ++c) o[(size_t)c * PP] = sum[c] * inv;
}

extern "C" void kernel_launch(void* const* d_in, const int* in_sizes, int n_in,
                              void* d_out, int out_size, void* d_ws, size_t ws_size,
                              hipStream_t stream) {
  const float* ft   = (const float*)d_in[0];   // (1, C*PH*PW, H, W) f32
  const float* rois = (const float*)d_in[1];   // (N, 5) f32
  float* out        = (float*)d_out;           // (N, C, P) f32
  const int N = in_sizes[1] / 5;

  dim3 grid((N + RPB - 1) / RPB, PP, 1);
  dfmb_psroi_align_kernel<<<grid, RPB, 0, stream>>>(ft, rois, out, N);
}